// QVConditionedGAT_79362405696008
// MI455X (gfx1250) — compile-verified
//
#include <hip/hip_runtime.h>
#include <math.h>

// Problem sizes (fixed by the reference)
#define Bn 8
#define Ln 20
#define Nn 2048
#define Dn 256

typedef __attribute__((ext_vector_type(2))) float v2f;
typedef __attribute__((ext_vector_type(8))) float v8f;
typedef int v4i __attribute__((vector_size(16)));

#define AS_GLOBAL __attribute__((address_space(1)))
#define AS_LDS __attribute__((address_space(3)))

// D = A(16x4) * B(4x16) + C, all fp32. wave32.
__device__ __forceinline__ v8f wmma4(v2f a, v2f b, v8f c) {
  return __builtin_amdgcn_wmma_f32_16x16x4_f32(false, a, false, b, (short)0, c,
                                               false, false);
}

// ---- async global->LDS 16B copy (per-lane addresses), ASYNCcnt tracked
__device__ __forceinline__ void async_ld_b128(const float* g, float* l) {
#if __has_builtin(__builtin_amdgcn_global_load_async_to_lds_b128)
  __builtin_amdgcn_global_load_async_to_lds_b128(
      (AS_GLOBAL v4i*)g, (AS_LDS v4i*)l, 0, 0);
#else
  unsigned loff = (unsigned)(uintptr_t)(AS_LDS void*)l;
  asm volatile("global_load_async_to_lds_b128 %0, %1, off" ::"v"(loff), "v"(g)
               : "memory");
#endif
}

__device__ __forceinline__ void wait_async2() {
#if __has_builtin(__builtin_amdgcn_s_wait_asynccnt)
  __builtin_amdgcn_s_wait_asynccnt(2);
#else
  asm volatile("s_wait_asynccnt 0x2" ::: "memory");
#endif
}

__device__ __forceinline__ void wait_async0() {
#if __has_builtin(__builtin_amdgcn_s_wait_asynccnt)
  __builtin_amdgcn_s_wait_asynccnt(0);
#else
  asm volatile("s_wait_asynccnt 0x0" ::: "memory");
#endif
}

// ---------------------------------------------------------------- init
__global__ __launch_bounds__(256) void k_init(float* key_vec) {
  int t = threadIdx.x;
  for (int i = t; i < Bn * Dn; i += 256) key_vec[i] = 0.f;
}

// ------------------------------------------------- key_vec = sum_n feat
__global__ __launch_bounds__(256) void k_featsum(const float* __restrict__ feat,
                                                 float* __restrict__ key_vec) {
  int b = blockIdx.y;
  int base = blockIdx.x * 128;  // 16 chunks of 128 rows
  int d = threadIdx.x;
  float s = 0.f;
  const float* p = feat + ((size_t)b * Nn + base) * Dn + d;
  for (int n = 0; n < 128; ++n) s += p[(size_t)n * Dn];
  atomicAdd(&key_vec[b * Dn + d], s);
}

// ---------------- additive-attention logits: one block per (b,l) token
__global__ __launch_bounds__(256) void k_softlogits(
    const float* __restrict__ ques, const float* __restrict__ key_vec,
    const float* __restrict__ Wq_a, const float* __restrict__ bq_a,
    const float* __restrict__ Wk_a, const float* __restrict__ bk_a,
    const float* __restrict__ ws_a, const float* __restrict__ bs_a,
    float* __restrict__ logits) {
  __shared__ float red[256];
  int b = blockIdx.y, l = blockIdx.x, j = threadIdx.x;
  const float* qrow = ques + ((size_t)b * Ln + l) * Dn;
  const float* kv = key_vec + b * Dn;
  float t = bq_a[j] + bk_a[j];
  for (int i = 0; i < Dn; ++i)
    t += qrow[i] * Wq_a[i * Dn + j] + kv[i] * Wk_a[i * Dn + j];
  red[j] = tanhf(t) * ws_a[j];
  __syncthreads();
  for (int s = 128; s > 0; s >>= 1) {
    if (j < s) red[j] += red[j + s];
    __syncthreads();
  }
  if (j == 0) logits[b * Ln + l] = red[0] + bs_a[0];
}

// --- softmax over L, ques_flat, then cvec[b,:] = ques_flat@W_fc[D:] + b_fc
__global__ __launch_bounds__(256) void k_combine(
    const float* __restrict__ ques, const float* __restrict__ logits,
    const float* __restrict__ W_fc, const float* __restrict__ b_fc,
    float* __restrict__ cvec) {
  __shared__ float qs[Dn];
  int b = blockIdx.x, d = threadIdx.x;
  const float* lg = logits + b * Ln;
  float mx = -1e30f;
  for (int l = 0; l < Ln; ++l) mx = fmaxf(mx, lg[l]);
  float se = 0.f;
  for (int l = 0; l < Ln; ++l) se += __expf(lg[l] - mx);
  float inv = 1.f / se;
  float qd = 0.f;
  for (int l = 0; l < Ln; ++l)
    qd += (__expf(lg[l] - mx) * inv) * ques[((size_t)b * Ln + l) * Dn + d];
  qs[d] = qd;
  __syncthreads();
  float cv = b_fc[d];
  for (int i = 0; i < Dn; ++i) cv += qs[i] * W_fc[(size_t)(Dn + i) * Dn + d];
  cvec[b * Dn + d] = cv;
}

// -------- feat_proj = feat @ W_fc[:D] + cvec     (WMMA f32 16x16x4)
// block = 8 waves: wave w -> (mtile = w>>2) x (nquarter = w&3, 64 cols)
__global__ __launch_bounds__(256) void k_gemm_proj(
    const float* __restrict__ feat, const float* __restrict__ W_fc,
    const float* __restrict__ cvec, float* __restrict__ fproj) {
  int b = blockIdx.y;
  int w = threadIdx.x >> 5, lane = threadIdx.x & 31;
  int h = lane >> 4;   // K-half within a WMMA step
  int ml = lane & 15;  // A-row / B-col within tile
  int m0 = blockIdx.x * 32 + (w >> 2) * 16;
  int nq = (w & 3) * 64;

  v8f acc[4] = {};
  const float* arow = feat + ((size_t)b * Nn + m0 + ml) * Dn;
  for (int k0 = 0; k0 < Dn; k0 += 4) {
    v2f a;
    a.x = arow[k0 + 2 * h];
    a.y = arow[k0 + 2 * h + 1];
    const float* wrow = W_fc + (size_t)(k0 + 2 * h) * Dn + nq + ml;
#pragma unroll
    for (int t = 0; t < 4; ++t) {
      v2f bb;
      bb.x = wrow[t * 16];
      bb.y = wrow[Dn + t * 16];
      acc[t] = wmma4(a, bb, acc[t]);
    }
  }
#pragma unroll
  for (int t = 0; t < 4; ++t) {
    int c = nq + t * 16 + ml;
    float cadd = cvec[b * Dn + c];
#pragma unroll
    for (int i = 0; i < 8; ++i) {
      int row = i + 8 * h;  // C layout: VGPR i -> rows i / i+8
      fproj[((size_t)b * Nn + m0 + row) * Dn + c] = acc[t][i] + cadd;
    }
  }
}

// -------- per-node scalar scores q_i, k_j (wave per row)
__global__ __launch_bounds__(256) void k_qk(const float* __restrict__ fproj,
                                            const float* __restrict__ w_q,
                                            const float* __restrict__ b_q,
                                            const float* __restrict__ w_k,
                                            const float* __restrict__ b_k,
                                            float* __restrict__ qf,
                                            float* __restrict__ kf) {
  int w = threadIdx.x >> 5, lane = threadIdx.x & 31;
  int gid = blockIdx.x * 8 + w;  // node row in [0, B*N)
  const float* fp = fproj + (size_t)gid * Dn;
  float sq = 0.f, sk = 0.f;
#pragma unroll
  for (int e = 0; e < 8; ++e) {
    int i = lane + 32 * e;
    float v = fp[i];
    sq += v * w_q[i];
    sk += v * w_k[i];
  }
#pragma unroll
  for (int off = 16; off > 0; off >>= 1) {
    sq += __shfl_xor(sq, off, 32);
    sk += __shfl_xor(sk, off, 32);
  }
  if (lane == 0) {
    qf[gid] = sq + b_q[0];
    kf[gid] = sk + b_k[0];
  }
}

// -------- fused mask + leaky-relu + online softmax + attn@V + residual
// grid (N/16, B); block 256 = 8 waves; wave w owns output cols [32w,32w+32)
// Adjacency (the single-use 134MB stream) is double-buffered into LDS with
// GLOBAL_LOAD_ASYNC_TO_LDS_B128: next chunk's HBM fetch overlaps the current
// chunk's softmax + WMMA work.
__global__ __launch_bounds__(256) void k_attn(
    const float* __restrict__ adj, const float* __restrict__ fproj,
    const float* __restrict__ qf, const float* __restrict__ kf,
    const float* __restrict__ feat, float* __restrict__ out) {
  __shared__ float adj_s[2][16 * 128];  // raw adjacency, double buffered
  __shared__ float s_tile[16 * 132];    // scores / probabilities (+pad)
  __shared__ float q_s[16], m_s[16], l_s[16], scale_s[16];

  int b = blockIdx.y;
  int i0 = blockIdx.x * 16;
  int tid = threadIdx.x;
  int w = tid >> 5, lane = tid & 31;
  int h = lane >> 4, ml = lane & 15;
  int r = tid >> 4, cbase = tid & 15;  // staging: 16 threads per row
  int lr = tid >> 5, lc = (tid & 31) * 4;  // async-copy mapping (b128)

  const float* adjb = adj + (size_t)b * Nn * Nn;
  const float* fpb = fproj + (size_t)b * Nn * Dn;

  if (tid < 16) {
    q_s[tid] = qf[b * Nn + i0 + tid];
    m_s[tid] = -1e30f;
    l_s[tid] = 0.f;
  }
  // kick off chunk 0 (two b128 issues per thread: rows lr and lr+8)
  async_ld_b128(adjb + (size_t)(i0 + lr) * Nn + lc, &adj_s[0][lr * 128 + lc]);
  async_ld_b128(adjb + (size_t)(i0 + lr + 8) * Nn + lc,
                &adj_s[0][(lr + 8) * 128 + lc]);
  __syncthreads();

  v8f acc0 = {}, acc1 = {};
  int cur = 0;

  for (int jc = 0; jc < Nn; jc += 128) {
    // ---- issue async fetch of next chunk, then drain current chunk
    if (jc + 128 < Nn) {
      const float* g = adjb + (size_t)(i0 + lr) * Nn + (jc + 128) + lc;
      async_ld_b128(g, &adj_s[cur ^ 1][lr * 128 + lc]);
      async_ld_b128(g + (size_t)8 * Nn, &adj_s[cur ^ 1][(lr + 8) * 128 + lc]);
      wait_async2();  // allow exactly the 2 just-issued to stay in flight
      __builtin_prefetch(fpb + (size_t)(jc + 128 + lr * 16) * Dn + lc * 2, 0,
                         0);
    } else {
      wait_async0();
    }
    __syncthreads();  // all waves' async writes to adj_s[cur] visible

    // ---- masked + leaky-relu scores from LDS adjacency
    float qr = q_s[r];
#pragma unroll
    for (int e = 0; e < 8; ++e) {
      int col = cbase + 16 * e;
      float a = adj_s[cur][r * 128 + col];
      float s = qr + kf[b * Nn + jc + col] + (1.f - a) * -1e9f;
      s = s > 0.f ? s : 0.01f * s;
      s_tile[r * 132 + col] = s;
    }
    __syncthreads();

    // ---- flash update: all 8 waves, 2 rows each, 16-lane shfl tree
    {
      int rr = 2 * w + (lane >> 4);
      int cl = lane & 15;
      float m_old = m_s[rr];
      float mn = m_old;
#pragma unroll
      for (int e = 0; e < 8; ++e)
        mn = fmaxf(mn, s_tile[rr * 132 + cl + 16 * e]);
#pragma unroll
      for (int off = 8; off > 0; off >>= 1)
        mn = fmaxf(mn, __shfl_xor(mn, off, 16));
      float ssum = 0.f;
#pragma unroll
      for (int e = 0; e < 8; ++e)
        ssum += __expf(s_tile[rr * 132 + cl + 16 * e] - mn);
#pragma unroll
      for (int off = 8; off > 0; off >>= 1) ssum += __shfl_xor(ssum, off, 16);
      if (cl == 0) {
        float sc = __expf(m_old - mn);
        m_s[rr] = mn;
        scale_s[rr] = sc;
        l_s[rr] = l_s[rr] * sc + ssum;
      }
    }
    __syncthreads();

    // ---- exponentiate tile in-place; rescale accumulators
#pragma unroll
    for (int e = 0; e < 8; ++e) {
      int col = cbase + 16 * e;
      s_tile[r * 132 + col] = __expf(s_tile[r * 132 + col] - m_s[r]);
    }
#pragma unroll
    for (int i = 0; i < 8; ++i) {
      float f = scale_s[i + 8 * h];
      acc0[i] *= f;
      acc1[i] *= f;
    }
    __syncthreads();

    // ---- P(16x128) @ V(128x32) via WMMA f32 16x16x4
    for (int k0 = 0; k0 < 128; k0 += 4) {
      v2f a;
      a.x = s_tile[ml * 132 + k0 + 2 * h];
      a.y = s_tile[ml * 132 + k0 + 2 * h + 1];
      const float* vrow = fpb + (size_t)(jc + k0 + 2 * h) * Dn + w * 32 + ml;
      v2f b0;
      b0.x = vrow[0];
      b0.y = vrow[Dn];
      acc0 = wmma4(a, b0, acc0);
      v2f b1;
      b1.x = vrow[16];
      b1.y = vrow[Dn + 16];
      acc1 = wmma4(a, b1, acc1);
    }
    __syncthreads();
    cur ^= 1;
  }

  // ---- normalize, residual, store
#pragma unroll
  for (int i = 0; i < 8; ++i) {
    int row = i + 8 * h;
    float invl = 1.f / l_s[row];
    size_t base = ((size_t)b * Nn + i0 + row) * Dn;
    int c0 = w * 32 + ml;
    out[base + c0] = acc0[i] * invl + feat[base + c0];
    out[base + c0 + 16] = acc1[i] * invl + feat[base + c0 + 16];
  }
}

// ---------------------------------------------------------------- launch
extern "C" void kernel_launch(void* const* d_in, const int* in_sizes, int n_in,
                              void* d_out, int out_size, void* d_ws,
                              size_t ws_size, hipStream_t stream) {
  const float* ques = (const float*)d_in[0];
  const float* feat = (const float*)d_in[1];
  const float* adj = (const float*)d_in[2];
  const float* Wq_a = (const float*)d_in[3];
  const float* bq_a = (const float*)d_in[4];
  const float* Wk_a = (const float*)d_in[5];
  const float* bk_a = (const float*)d_in[6];
  const float* ws_a = (const float*)d_in[7];
  const float* bs_a = (const float*)d_in[8];
  const float* W_fc = (const float*)d_in[9];
  const float* b_fc = (const float*)d_in[10];
  const float* w_q = (const float*)d_in[11];
  const float* b_q = (const float*)d_in[12];
  const float* w_k = (const float*)d_in[13];
  const float* b_k = (const float*)d_in[14];
  float* out = (float*)d_out;

  float* wsf = (float*)d_ws;
  float* key_vec = wsf;                   // 2048
  float* logits = wsf + 2048;             // 160 (padded to 256)
  float* cvec = wsf + 2048 + 256;         // 2048
  float* qf = wsf + 4352;                 // 16384
  float* kf = wsf + 4352 + 16384;         // 16384
  float* fproj = wsf + 4352 + 2 * 16384;  // B*N*D = 4194304

  k_init<<<1, 256, 0, stream>>>(key_vec);
  k_featsum<<<dim3(16, Bn), 256, 0, stream>>>(feat, key_vec);
  k_softlogits<<<dim3(Ln, Bn), 256, 0, stream>>>(ques, key_vec, Wq_a, bq_a,
                                                 Wk_a, bk_a, ws_a, bs_a,
                                                 logits);
  k_combine<<<Bn, 256, 0, stream>>>(ques, logits, W_fc, b_fc, cvec);
  k_gemm_proj<<<dim3(Nn / 32, Bn), 256, 0, stream>>>(feat, W_fc, cvec, fproj);
  k_qk<<<(Bn * Nn) / 8, 256, 0, stream>>>(fproj, w_q, b_q, w_k, b_k, qf, kf);
  k_attn<<<dim3(Nn / 16, Bn), 256, 0, stream>>>(adj, fproj, qf, kf, feat, out);
}